// S4Model_7438883357296
// MI455X (gfx1250) — compile-verified
//
#include <hip/hip_runtime.h>
#include <cstdint>

typedef __attribute__((ext_vector_type(2))) float v2f;
typedef __attribute__((ext_vector_type(8))) float v8f;

constexpr int BSZ = 16, T = 2048, DIN = 256, H = 512;
constexpr int M1  = BSZ * T;            // 32768 rows
constexpr int KC  = 16;                 // GEMM K-chunk (4 WMMA k-steps)
constexpr int KP  = 20;                 // padded LDS row stride (bank spread)
constexpr int SCAN_CHUNK = 64;          // timesteps staged per iteration

// ---------------------------------------------------------------------------
// CDNA5 async global->LDS copy helpers (ASYNCcnt-tracked, GV addressing mode).
// LDS destination address = 32-bit offset of the addrspace(3) pointer.
// ---------------------------------------------------------------------------
#define LDS_AS __attribute__((address_space(3)))

__device__ __forceinline__ uint32_t lds_off_u32(const void* p) {
    return (uint32_t)(uintptr_t)(LDS_AS const void*)p;
}
__device__ __forceinline__ void async_g2l_b128(void* lds_dst, const void* gsrc) {
    asm volatile("global_load_async_to_lds_b128 %0, %1, off"
                 :: "v"(lds_off_u32(lds_dst)), "v"((uint64_t)(uintptr_t)gsrc)
                 : "memory");
}
__device__ __forceinline__ void async_g2l_b64(void* lds_dst, const void* gsrc) {
    asm volatile("global_load_async_to_lds_b64 %0, %1, off"
                 :: "v"(lds_off_u32(lds_dst)), "v"((uint64_t)(uintptr_t)gsrc)
                 : "memory");
}
__device__ __forceinline__ void async_g2l_b32(void* lds_dst, const void* gsrc) {
    asm volatile("global_load_async_to_lds_b32 %0, %1, off"
                 :: "v"(lds_off_u32(lds_dst)), "v"((uint64_t)(uintptr_t)gsrc)
                 : "memory");
}
__device__ __forceinline__ void wait_async0() {
    asm volatile("s_wait_asynccnt 0" ::: "memory");
}

// ---------------------------------------------------------------------------
// Kernel 1: xh[M1,H] = x[M1,DIN] @ W[DIN,H] + bias[H]   (f32 WMMA 16x16x4)
// Block tile 128(M) x 64(N), 8 waves, each wave 32x32 (2x2 WMMA accumulators).
// K chunked by 16, double-buffered async LDS staging; B staged transposed
// ([n][k]) so both fragments are single ds_load_b64s.
// ---------------------------------------------------------------------------
__global__ __launch_bounds__(256) void proj_gemm(
    const float* __restrict__ x, const float* __restrict__ W,
    const float* __restrict__ bias, float* __restrict__ xh)
{
    __shared__ float As[2][128][KP];    // [m][k]
    __shared__ float Bs[2][64][KP];     // [n][k] (transposed)

    const int tid  = threadIdx.x;
    const int lane = tid & 31, wave = tid >> 5;
    const int wr   = wave & 3, wc = wave >> 2;      // wave grid 4(M) x 2(N)
    const int mblk = blockIdx.x * 128, nblk = blockIdx.y * 64;
    const int mr   = lane & 15, hi = lane >> 4;

    v8f acc[2][2] = {};

    auto stage = [&](int buf, int k0) {
        // A tile: 128 x 16 floats, b128 granules: 512 granules / 256 threads
#pragma unroll
        for (int it = 0; it < 2; ++it) {
            const int e = tid + it * 256;
            const int r = e >> 2, c4 = (e & 3) * 4;
            async_g2l_b128(&As[buf][r][c4],
                           x + (size_t)(mblk + r) * DIN + k0 + c4);
        }
        // B tile (transposed into LDS): 16 x 64 elems, b32 granules
#pragma unroll
        for (int it = 0; it < 4; ++it) {
            const int e = tid + it * 256;
            const int kk = e >> 6, n = e & 63;
            async_g2l_b32(&Bs[buf][n][kk],
                          W + (size_t)(k0 + kk) * H + nblk + n);
        }
    };

    stage(0, 0);
    int buf = 0;
    for (int k0 = 0; k0 < DIN; k0 += KC) {
        wait_async0();
        __syncthreads();
        if (k0 + KC < DIN) stage(buf ^ 1, k0 + KC);
#pragma unroll
        for (int kk = 0; kk < KC; kk += 4) {
            v2f a[2], b[2];
#pragma unroll
            for (int mt = 0; mt < 2; ++mt)
                a[mt] = *(const v2f*)&As[buf][wr * 32 + mt * 16 + mr][kk + 2 * hi];
#pragma unroll
            for (int nt = 0; nt < 2; ++nt)
                b[nt] = *(const v2f*)&Bs[buf][wc * 32 + nt * 16 + mr][kk + 2 * hi];
#pragma unroll
            for (int mt = 0; mt < 2; ++mt)
#pragma unroll
                for (int nt = 0; nt < 2; ++nt)
                    acc[mt][nt] = __builtin_amdgcn_wmma_f32_16x16x4_f32(
                        false, a[mt], false, b[nt], (short)0, acc[mt][nt],
                        false, false);
        }
        __syncthreads();
        buf ^= 1;
    }

#pragma unroll
    for (int mt = 0; mt < 2; ++mt)
#pragma unroll
        for (int nt = 0; nt < 2; ++nt) {
            const int col = nblk + wc * 32 + nt * 16 + mr;
            const float bv = bias[col];
#pragma unroll
            for (int v = 0; v < 8; ++v) {
                const int row = mblk + wr * 32 + mt * 16 + v + 8 * hi;
                xh[(size_t)row * H + col] = acc[mt][nt][v] + bv;
            }
        }
}

// ---------------------------------------------------------------------------
// Kernel 2: diagonal complex SSM scan.
//   conv[b,t,c] = sum_s B[s,c]*w[t,s],  w[t,s] = A[s]*w[t-1,s] + xh[b,t,c]
//   (+ h0[c] at t==0).  One wave per channel c: 32 lanes x 16 states/lane.
// xh streamed through double-buffered LDS via async-to-LDS b64 copies.
// ---------------------------------------------------------------------------
__global__ __launch_bounds__(512) void scan_kernel(
    const float* __restrict__ xh, const float* __restrict__ Ar,
    const float* __restrict__ Ai, const float* __restrict__ Bm,
    const float* __restrict__ h0,
    float* __restrict__ conv_re, float* __restrict__ conv_im)
{
    const int b   = blockIdx.y;
    const int tid = threadIdx.x;
    const int jl  = tid >> 5;                 // channel within block: 0..15
    const int sg  = tid & 31;                 // lane: state group
    const int j   = blockIdx.x * 16 + jl;     // global channel
    const int s0  = sg * 16;                  // 16 states per lane

    float ar[16], ai[16], bw[16], wr[16], wi[16];
#pragma unroll
    for (int u = 0; u < 16; ++u) {
        ar[u] = Ar[s0 + u];
        ai[u] = Ai[s0 + u];
        bw[u] = Bm[(size_t)(s0 + u) * H + j];
        wr[u] = 0.f; wi[u] = 0.f;
    }
    const float h0j = h0[j];

    __shared__ float xs[2][SCAN_CHUNK][16];   // 2 x 64 timesteps x 16 channels

    const float* xbase = xh + (size_t)b * T * H + blockIdx.x * 16;

    auto stage = [&](int buf, int t0) {
        // 64*16 = 1024 floats, b64 granules: 512 granules / 512 threads
        const int tt = tid >> 3, c2 = (tid & 7) * 2;
        async_g2l_b64(&xs[buf][tt][c2],
                      xbase + (size_t)(t0 + tt) * H + c2);
    };

    stage(0, 0);
    int buf = 0;
    for (int t0 = 0; t0 < T; t0 += SCAN_CHUNK) {
        wait_async0();
        __syncthreads();
        if (t0 + SCAN_CHUNK < T) stage(buf ^ 1, t0 + SCAN_CHUNK);

        for (int dt = 0; dt < SCAN_CHUNK; ++dt) {
            const int t = t0 + dt;
            const float xv = xs[buf][dt][jl];  // broadcast within wave
            float pr = 0.f, pi = 0.f;
#pragma unroll
            for (int u = 0; u < 16; ++u) {
                const float nr = ar[u] * wr[u] - ai[u] * wi[u] + xv;
                const float ni = ar[u] * wi[u] + ai[u] * wr[u];
                wr[u] = nr; wi[u] = ni;
                pr += bw[u] * nr;
                pi += bw[u] * ni;
            }
#pragma unroll
            for (int m = 1; m < 32; m <<= 1) {
                pr += __shfl_xor(pr, m, 32);
                pi += __shfl_xor(pi, m, 32);
            }
            if (sg == 0) {
                if (t == 0) pr += h0j;
                const size_t idx = ((size_t)b * T + t) * H + j;
                conv_re[idx] = pr;
                conv_im[idx] = pi;
            }
        }
        __syncthreads();
        buf ^= 1;
    }
}

// ---------------------------------------------------------------------------
// Kernel 3: y = conv @ C  (complex x real -> paired f32 WMMA accumulators
// sharing the C fragment), interleaved complex64 store to d_out.
// Same tiling/async-staging scheme as proj_gemm, with re/im A planes.
// ---------------------------------------------------------------------------
__global__ __launch_bounds__(256) void out_gemm(
    const float* __restrict__ cre, const float* __restrict__ cim,
    const float* __restrict__ C, float* __restrict__ out)
{
    __shared__ float Ars[2][128][KP];
    __shared__ float Ais[2][128][KP];
    __shared__ float Bs [2][64][KP];

    const int tid  = threadIdx.x;
    const int lane = tid & 31, wave = tid >> 5;
    const int wr   = wave & 3, wc = wave >> 2;
    const int mblk = blockIdx.x * 128, nblk = blockIdx.y * 64;
    const int mr   = lane & 15, hi = lane >> 4;

    v8f accr[2][2] = {}, acci[2][2] = {};

    auto stage = [&](int buf, int k0) {
#pragma unroll
        for (int it = 0; it < 2; ++it) {
            const int e = tid + it * 256;
            const int r = e >> 2, c4 = (e & 3) * 4;
            async_g2l_b128(&Ars[buf][r][c4],
                           cre + (size_t)(mblk + r) * H + k0 + c4);
            async_g2l_b128(&Ais[buf][r][c4],
                           cim + (size_t)(mblk + r) * H + k0 + c4);
        }
#pragma unroll
        for (int it = 0; it < 4; ++it) {
            const int e = tid + it * 256;
            const int kk = e >> 6, n = e & 63;
            async_g2l_b32(&Bs[buf][n][kk],
                          C + (size_t)(k0 + kk) * H + nblk + n);
        }
    };

    stage(0, 0);
    int buf = 0;
    for (int k0 = 0; k0 < H; k0 += KC) {
        wait_async0();
        __syncthreads();
        if (k0 + KC < H) stage(buf ^ 1, k0 + KC);
#pragma unroll
        for (int kk = 0; kk < KC; kk += 4) {
            v2f arf[2], aif[2], b[2];
#pragma unroll
            for (int mt = 0; mt < 2; ++mt) {
                arf[mt] = *(const v2f*)&Ars[buf][wr * 32 + mt * 16 + mr][kk + 2 * hi];
                aif[mt] = *(const v2f*)&Ais[buf][wr * 32 + mt * 16 + mr][kk + 2 * hi];
            }
#pragma unroll
            for (int nt = 0; nt < 2; ++nt)
                b[nt] = *(const v2f*)&Bs[buf][wc * 32 + nt * 16 + mr][kk + 2 * hi];
#pragma unroll
            for (int mt = 0; mt < 2; ++mt)
#pragma unroll
                for (int nt = 0; nt < 2; ++nt) {
                    accr[mt][nt] = __builtin_amdgcn_wmma_f32_16x16x4_f32(
                        false, arf[mt], false, b[nt], (short)0, accr[mt][nt],
                        false, false);
                    acci[mt][nt] = __builtin_amdgcn_wmma_f32_16x16x4_f32(
                        false, aif[mt], false, b[nt], (short)0, acci[mt][nt],
                        false, false);
                }
        }
        __syncthreads();
        buf ^= 1;
    }

#pragma unroll
    for (int mt = 0; mt < 2; ++mt)
#pragma unroll
        for (int nt = 0; nt < 2; ++nt) {
            const int col = nblk + wc * 32 + nt * 16 + mr;
#pragma unroll
            for (int v = 0; v < 8; ++v) {
                const int row = mblk + wr * 32 + mt * 16 + v + 8 * hi;
                const size_t idx = ((size_t)row * H + col) * 2;
                out[idx]     = accr[mt][nt][v];
                out[idx + 1] = acci[mt][nt][v];
            }
        }
}

// ---------------------------------------------------------------------------
extern "C" void kernel_launch(void* const* d_in, const int* in_sizes, int n_in,
                              void* d_out, int out_size, void* d_ws, size_t ws_size,
                              hipStream_t stream)
{
    const float* x    = (const float*)d_in[0];
    const float* W    = (const float*)d_in[1];
    const float* bias = (const float*)d_in[2];
    const float* Ar   = (const float*)d_in[3];
    const float* Ai   = (const float*)d_in[4];
    const float* Bm   = (const float*)d_in[5];
    const float* C    = (const float*)d_in[6];
    const float* h0   = (const float*)d_in[7];
    float* out = (float*)d_out;

    float* xh      = (float*)d_ws;                       // [M1, H]
    float* conv_re = xh      + (size_t)M1 * H;           // [M1, H]
    float* conv_im = conv_re + (size_t)M1 * H;           // [M1, H]

    proj_gemm <<<dim3(M1 / 128, H / 64), 256, 0, stream>>>(x, W, bias, xh);
    scan_kernel<<<dim3(H / 16, BSZ),     512, 0, stream>>>(xh, Ar, Ai, Bm, h0,
                                                           conv_re, conv_im);
    out_gemm  <<<dim3(M1 / 128, H / 64), 256, 0, stream>>>(conv_re, conv_im, C, out);
}